// H2GCNConv_16604343566796
// MI455X (gfx1250) — compile-verified
//
#include <hip/hip_runtime.h>

// ---------------- CDNA5 async global->LDS helpers ----------------
// gfx1250: VDST = VGPR holding 32-bit LDS byte offset, VADDR = 64-bit global
// address, 'off' = no SADDR. Tracked by ASYNCcnt. Generic pointers to
// __shared__ carry the LDS offset in their low 32 bits, so a truncating cast
// yields the correct VDST value (same base ds_load_b64 uses).
__device__ __forceinline__ void asyncPairToLds(const int2* g, int2* l) {
  unsigned loff = (unsigned)(size_t)l;
  asm volatile("global_load_async_to_lds_b64 %0, %1, off"
               :: "v"(loff), "v"(g) : "memory");
}

#if __has_builtin(__builtin_amdgcn_s_wait_asynccnt)
#define ASYNC_WAIT(N)                                \
  do {                                               \
    __builtin_amdgcn_s_wait_asynccnt(N);             \
    asm volatile("" ::: "memory");                   \
  } while (0)
#else
#define ASYNC_WAIT(N) asm volatile("s_wait_asynccnt %0" :: "n"(N) : "memory")
#endif

// ---------------- utility kernels ----------------
__global__ void zero_i32_kernel(int* __restrict__ p, int n) {
  int i = blockIdx.x * blockDim.x + threadIdx.x;
  if (i < n) p[i] = 0;
}

__global__ void zero_f32x4_kernel(float4* __restrict__ p, int n4) {
  int i = blockIdx.x * blockDim.x + threadIdx.x;
  if (i < n4) p[i] = make_float4(0.f, 0.f, 0.f, 0.f);
}

// ---------------- phase 1: row histogram ----------------
__global__ void hist_kernel(const int* __restrict__ row, int e, int* __restrict__ cnt) {
  int i = blockIdx.x * blockDim.x + threadIdx.x;
  if (i < e) atomicAdd(&cnt[row[i]], 1);
}

// ---------------- phase 2: exclusive scan (single workgroup, chunked) ----------------
__global__ void __launch_bounds__(1024) scan_kernel(const int* __restrict__ cnt,
                                                    int* __restrict__ off,
                                                    int* __restrict__ cur, int n) {
  __shared__ int buf[2][1024];
  __shared__ int carry_s;
  const int t = threadIdx.x;
  if (t == 0) carry_s = 0;
  __syncthreads();
  for (int base = 0; base < n; base += 1024) {
    const int i = base + t;
    const int v = (i < n) ? cnt[i] : 0;
    buf[0][t] = v;
    __syncthreads();
    int s = 0;
    #pragma unroll
    for (int d = 1; d < 1024; d <<= 1) {
      int val = buf[s][t];
      if (t >= d) val += buf[s][t - d];
      buf[s ^ 1][t] = val;   // other buffer: no race with reads of buf[s]
      s ^= 1;
      __syncthreads();
    }
    const int incl = buf[s][t];
    const int carry = carry_s;
    __syncthreads();                    // all threads read carry before update
    if (i < n) {
      const int o = carry + incl - v;   // exclusive
      off[i] = o;
      cur[i] = o;
    }
    if (t == 1023) carry_s = carry + incl;
    __syncthreads();
  }
  if (t == 0) off[n] = carry_s;
}

// ---------------- phase 3: scatter edges into row-sorted (col,val) pairs ----------------
__global__ void scatter_kernel(const int* __restrict__ row, const int* __restrict__ col,
                               const float* __restrict__ val, int e,
                               int* __restrict__ cur, int2* __restrict__ pairs) {
  int i = blockIdx.x * blockDim.x + threadIdx.x;
  if (i < e) {
    const int r = row[i];
    const int p = atomicAdd(&cur[r], 1);
    pairs[p] = make_int2(col[i], __float_as_int(val[i]));
  }
}

// ---------------- phase 4: wave-per-row gather/accumulate, async-staged edge stream ----------------
__global__ void __launch_bounds__(256) spmm_rows_kernel(
    const float* __restrict__ x, const int* __restrict__ off,
    const int2* __restrict__ pairs, float* __restrict__ out, int n, int colOff) {
  __shared__ int2 stage[8][2][32];          // per-wave double buffer of 32 edges
  const int wid  = threadIdx.x >> 5;
  const int lane = threadIdx.x & 31;
  const int r = blockIdx.x * 8 + wid;
  if (r >= n) return;                        // whole-wave uniform exit, no block barriers here

  const int beg = off[r];
  const int end = off[r + 1];
  const int cnt = end - beg;
  float4 acc = make_float4(0.f, 0.f, 0.f, 0.f);
  const float4* __restrict__ x4 = (const float4*)x;

  const int nchunks = (cnt + 31) >> 5;
  if (nchunks > 0) {
    {   // prime chunk 0 (clamped indices -> always in-bounds loads)
      int e0 = beg + lane; if (e0 >= end) e0 = end - 1;
      asyncPairToLds(&pairs[e0], &stage[wid][0][lane]);
    }
    for (int k = 0; k < nchunks; ++k) {
      if (k + 1 < nchunks) {
        int e1 = beg + ((k + 1) << 5) + lane; if (e1 >= end) e1 = end - 1;
        asyncPairToLds(&pairs[e1], &stage[wid][(k + 1) & 1][lane]);
        ASYNC_WAIT(1);                       // chunk k complete (in-order completion)
      } else {
        ASYNC_WAIT(0);
      }
      const int rem = cnt - (k << 5);
      const int m = rem < 32 ? rem : 32;
      const int2* buf = stage[wid][k & 1];
      #pragma unroll 4
      for (int j = 0; j < m; ++j) {
        const int2 pr = buf[j];              // ds_load_b64, broadcast to all lanes
        const float v = __int_as_float(pr.y);
        const float4 t = x4[(size_t)pr.x * 32 + lane];  // coalesced 512B row gather
        acc.x += v * t.x; acc.y += v * t.y;
        acc.z += v * t.z; acc.w += v * t.w;
      }
    }
  }
  ((float4*)(out + (size_t)r * 256 + colOff))[lane] = acc;  // no atomics
}

// ---------------- fallback: edge-parallel atomic SpMM (only if ws too small) ----------------
__global__ void __launch_bounds__(256) spmm_atomic_kernel(
    const float* __restrict__ x, const int* __restrict__ row, const int* __restrict__ col,
    const float* __restrict__ val, int e, float* __restrict__ out, int colOff) {
  const int wave = (blockIdx.x * blockDim.x + threadIdx.x) >> 5;
  const int lane = threadIdx.x & 31;
  if (wave >= e) return;
  const int r = row[wave];
  const int c = col[wave];
  const float v = val[wave];
  const float4 t = ((const float4*)x)[(size_t)c * 32 + lane];
  float* o = out + (size_t)r * 256 + colOff + lane * 4;
  unsafeAtomicAdd(o + 0, v * t.x);
  unsafeAtomicAdd(o + 1, v * t.y);
  unsafeAtomicAdd(o + 2, v * t.z);
  unsafeAtomicAdd(o + 3, v * t.w);
}

// ---------------- host launch ----------------
extern "C" void kernel_launch(void* const* d_in, const int* in_sizes, int n_in,
                              void* d_out, int out_size, void* d_ws, size_t ws_size,
                              hipStream_t stream) {
  const float* x    = (const float*)d_in[0];
  const int*   row1 = (const int*)d_in[1];
  const int*   col1 = (const int*)d_in[2];
  const float* val1 = (const float*)d_in[3];
  const int*   row2 = (const int*)d_in[4];
  const int*   col2 = (const int*)d_in[5];
  const float* val2 = (const float*)d_in[6];
  float* out = (float*)d_out;

  const int n  = in_sizes[0] / 128;   // 50000
  const int e1 = in_sizes[1];         // 800000
  const int e2 = in_sizes[4];         // 1600000

  // carve workspace (256B aligned segments)
  char* w = (char*)d_ws;
  size_t cursor = 0;
  auto carve = [&](size_t bytes) -> char* {
    char* p = w + cursor;
    cursor += (bytes + 255) & ~(size_t)255;
    return p;
  };
  int*  cnt1   = (int*)carve((size_t)n * 4);
  int*  cnt2   = (int*)carve((size_t)n * 4);
  int*  off1   = (int*)carve(((size_t)n + 1) * 4);
  int*  off2   = (int*)carve(((size_t)n + 1) * 4);
  int*  cur1   = (int*)carve((size_t)n * 4);
  int*  cur2   = (int*)carve((size_t)n * 4);
  int2* pairs1 = (int2*)carve((size_t)e1 * 8);
  int2* pairs2 = (int2*)carve((size_t)e2 * 8);

  const int B = 256;
  if (cursor <= ws_size) {
    // ---- sorted-CSR path: no float atomics ----
    zero_i32_kernel<<<(n + B - 1) / B, B, 0, stream>>>(cnt1, n);
    zero_i32_kernel<<<(n + B - 1) / B, B, 0, stream>>>(cnt2, n);
    hist_kernel<<<(e1 + B - 1) / B, B, 0, stream>>>(row1, e1, cnt1);
    hist_kernel<<<(e2 + B - 1) / B, B, 0, stream>>>(row2, e2, cnt2);
    scan_kernel<<<1, 1024, 0, stream>>>(cnt1, off1, cur1, n);
    scan_kernel<<<1, 1024, 0, stream>>>(cnt2, off2, cur2, n);
    scatter_kernel<<<(e1 + B - 1) / B, B, 0, stream>>>(row1, col1, val1, e1, cur1, pairs1);
    scatter_kernel<<<(e2 + B - 1) / B, B, 0, stream>>>(row2, col2, val2, e2, cur2, pairs2);
    const int rowBlocks = (n + 7) / 8;     // 8 waves (rows) per block
    spmm_rows_kernel<<<rowBlocks, B, 0, stream>>>(x, off1, pairs1, out, n, 0);
    spmm_rows_kernel<<<rowBlocks, B, 0, stream>>>(x, off2, pairs2, out, n, 128);
  } else {
    // ---- fallback: atomic scatter-add ----
    const int n4 = out_size / 4;
    zero_f32x4_kernel<<<(n4 + B - 1) / B, B, 0, stream>>>((float4*)out, n4);
    spmm_atomic_kernel<<<(int)(((size_t)e1 * 32 + B - 1) / B), B, 0, stream>>>(
        x, row1, col1, val1, e1, out, 0);
    spmm_atomic_kernel<<<(int)(((size_t)e2 * 32 + B - 1) / B), B, 0, stream>>>(
        x, row2, col2, val2, e2, out, 128);
  }
}